// QuanvolutionEstimator_65481071401801
// MI455X (gfx1250) — compile-verified
//
#include <hip/hip_runtime.h>
#include <hip/hip_bf16.h>
#include <math.h>

typedef __attribute__((ext_vector_type(2))) float v2f;
typedef __attribute__((ext_vector_type(8))) float v8f;

#define KDIM 784          // 28*28 pixels = folded K dimension
#define NOUT 10
#define NPAD 16
#define WAVES_PER_BLOCK 8
#define ROWS_PER_WAVE 16
#define ROWS_PER_BLOCK (WAVES_PER_BLOCK * ROWS_PER_WAVE)   // 128

// ---------------------------------------------------------------------------
// Kernel 1: fold conv(2x2,s2) + per-patch 4x4 linear + 1568->10 linear into a
// single 784->10 weight (stored K-major, N padded to 16 with zeros) + bias
// (also zero-padded to 16).
// ---------------------------------------------------------------------------
__global__ void quanv_fold_weights_kernel(const float* __restrict__ conv_w,
                                          const float* __restrict__ conv_b,
                                          const float* __restrict__ q_w,
                                          const float* __restrict__ lin_w,
                                          const float* __restrict__ lin_b,
                                          float* __restrict__ wsB,
                                          float* __restrict__ wsBias) {
  const int t = threadIdx.x;
  for (int idx = t; idx < KDIM * NPAD; idx += blockDim.x) {
    const int k = idx / NPAD;   // pixel index h*28 + w
    const int n = idx % NPAD;   // output logit (padded)
    float v = 0.f;
    if (n < NOUT) {
      const int h = k / 28, w = k % 28;
      const int r = h >> 1, dr = h & 1, c = w >> 1, dc = w & 1;
      const int p = r * 14 + c;       // patch index (r,c order)
      const int i = dr * 2 + dc;      // in-patch index (row-major 2x2)
      const float* lw = lin_w + n * 1568;
#pragma unroll
      for (int o = 0; o < 4; ++o) {
        v += conv_w[o * 4 + i] * lw[o * 196 + p];        // conv branch
        v += q_w[o * 4 + i]    * lw[784 + p * 4 + o];    // quantum branch
      }
    }
    wsB[idx] = v;   // layout: wsB[k*16 + n]  (B matrix K-major rows of 16)
  }
  if (t < NPAD) {
    float b = 0.f;
    if (t < NOUT) {
      b = lin_b[t];
      const float* lw = lin_w + t * 1568;
#pragma unroll
      for (int o = 0; o < 4; ++o) {
        float s = 0.f;
        for (int p = 0; p < 196; ++p) s += lw[o * 196 + p];
        b += conv_b[o] * s;
      }
    }
    wsBias[t] = b;  // zero for t >= NOUT -> unconditional add is safe
  }
}

// ---------------------------------------------------------------------------
// Kernel 2: logits = x(B,784) @ W2(784,16pad) + bias, then row log_softmax.
// One wave per 16-row batch tile, V_WMMA_F32_16X16X4_F32 over K=784.
// ---------------------------------------------------------------------------
__global__ __launch_bounds__(256)
void quanv_gemm_logsoftmax_kernel(const float* __restrict__ x,
                                  const float* __restrict__ wsB,
                                  const float* __restrict__ wsBias,
                                  float* __restrict__ out) {
  __shared__ float sB[KDIM * NPAD];               // 49 KB: folded weights
  __shared__ float sLog[ROWS_PER_BLOCK * NPAD];   // 8 KB:  logits staging
  __shared__ float sBias[NPAD];

  const int tid = threadIdx.x;

  // Stage folded weight + bias into LDS once per block.
  for (int i = tid; i < KDIM * NPAD; i += 256) sB[i] = wsB[i];
  if (tid < NPAD) sBias[tid] = wsBias[tid];
  __syncthreads();

  const int wave = tid >> 5;
  const int lane = tid & 31;
  const int half = lane >> 4;   // lane half selects K sub-pair (ISA A/B striping)
  const int l16  = lane & 15;

  const int rowBase = (blockIdx.x * WAVES_PER_BLOCK + wave) * ROWS_PER_WAVE;
  const float* xrow = x + (size_t)(rowBase + l16) * KDIM;  // A row for this lane pair

  v8f acc = {};
  for (int k = 0; k < KDIM; k += 4) {
    const int kk = k + half * 2;
    // A fragment (16x4 f32): lanes 0-15 hold K=k..k+1, lanes 16-31 K=k+2..k+3.
    v2f a;
    a.x = xrow[kk];
    a.y = xrow[kk + 1];
    // B fragment (4x16 f32): VGPR0 rows {k, k+2}, VGPR1 rows {k+1, k+3},
    // N striped across lanes 0-15 (same row striping as the C/D layout).
    v2f b;
    b.x = sB[kk * NPAD + l16];
    b.y = sB[(kk + 1) * NPAD + l16];
    // Stream hint: pull the next 128B of this row toward the WGP.
    __builtin_prefetch(&xrow[kk + 32], 0, 1);
    // D = A*B + C  (8 args: neg_a, A, neg_b, B, c_mod, C, reuse_a, reuse_b)
    acc = __builtin_amdgcn_wmma_f32_16x16x4_f32(false, a, false, b,
                                                (short)0, acc, false, false);
  }

  // Spill C tile to LDS per ISA layout: VGPR j -> M = j + 8*half, N = lane&15.
  // sBias is zero-padded, so add unconditionally (no EXEC divergence).
  const float biasv = sBias[l16];
#pragma unroll
  for (int j = 0; j < 8; ++j) {
    const int rlocal = wave * ROWS_PER_WAVE + half * 8 + j;
    sLog[rlocal * NPAD + l16] = acc[j] + biasv;
  }
  __syncthreads();

  // Per-row log_softmax over the 10 valid logits.
  for (int r = tid; r < ROWS_PER_BLOCK; r += 256) {
    const float* row = &sLog[r * NPAD];
    float m = row[0];
#pragma unroll
    for (int n = 1; n < NOUT; ++n) m = fmaxf(m, row[n]);
    float s = 0.f;
#pragma unroll
    for (int n = 0; n < NOUT; ++n) s += __expf(row[n] - m);
    const float lse = m + __logf(s);
    float* o = out + (size_t)(blockIdx.x * ROWS_PER_BLOCK + r) * NOUT;
#pragma unroll
    for (int n = 0; n < NOUT; ++n) o[n] = row[n] - lse;
  }
}

// ---------------------------------------------------------------------------
extern "C" void kernel_launch(void* const* d_in, const int* in_sizes, int n_in,
                              void* d_out, int out_size, void* d_ws, size_t ws_size,
                              hipStream_t stream) {
  (void)n_in; (void)out_size; (void)ws_size;
  const float* x      = (const float*)d_in[0];  // (B,1,28,28)
  const float* conv_w = (const float*)d_in[1];  // (4,1,2,2)
  const float* conv_b = (const float*)d_in[2];  // (4,)
  const float* q_w    = (const float*)d_in[3];  // (4,4)
  const float* lin_w  = (const float*)d_in[4];  // (10,1568)
  const float* lin_b  = (const float*)d_in[5];  // (10,)
  float* out = (float*)d_out;                   // (B,10)

  float* wsB    = (float*)d_ws;                 // 784*16 floats
  float* wsBias = wsB + KDIM * NPAD;            // 16 floats

  quanv_fold_weights_kernel<<<1, 256, 0, stream>>>(conv_w, conv_b, q_w,
                                                   lin_w, lin_b, wsB, wsBias);

  const int B = in_sizes[0] / KDIM;             // 32768
  const int grid = B / ROWS_PER_BLOCK;          // 256 blocks of 8 waves
  quanv_gemm_logsoftmax_kernel<<<grid, 256, 0, stream>>>(x, wsB, wsBias, out);
}